// ScalingAndSquaring_58720792871013
// MI455X (gfx1250) — compile-verified
//
#include <hip/hip_runtime.h>

// Scaling-and-squaring (7 steps) of a (2,3,160,160,160) fp32 velocity field.
// Memory/gather-bound (no matmul structure -> WMMA inapplicable; ~1 flop/byte).
// Roofline: ~1.6 GB of traffic across 8 launches; the 98.3 MB field is L2-
// resident (192 MB L2), so steps run mostly out of L2 -> minimize gather
// instruction count. AoS repack [b][d][h][w][3] turns each trilinear corner
// into one contiguous 12B (b96) read. Ping-pong d_ws <-> d_out; final step
// emits SoA. gfx1250 path: async global->LDS staging of the coalesced
// own-point read (global_load_async_to_lds_b32 + s_wait_asynccnt, ASYNCcnt).

#define DD 160
#define HH 160
#define WW 160
#define DHW (DD * HH * WW)        // 4,096,000
#define NBATCH 2
#define NPTS (NBATCH * DHW)       // 8,192,000
#define TPB 256
#define NBLK (NPTS / TPB)         // 32,000 (exact -> no tail threads)

__device__ __forceinline__ float unnorm_clip(float u) {
    // sg = 2*((g+v)/159 - 0.5);  i = ((sg+1)*160 - 1)*0.5;  clip to [0,159]
    float sg = 2.0f * (u * (1.0f / 159.0f) - 0.5f);
    float i  = ((sg + 1.0f) * 160.0f - 1.0f) * 0.5f;
    return fminf(fmaxf(i, 0.0f), 159.0f);
}

extern "C" __global__ void __launch_bounds__(TPB)
saq_step(const float* __restrict__ src, float* __restrict__ dst, int soa_out) {
    __shared__ float sm[TPB * 3];
    const int t   = threadIdx.x;
    const int idx = blockIdx.x * TPB + t;
    const float* gp = src + (size_t)idx * 3u;

    float v0, v1, v2;
#if defined(__gfx1250__) && __has_builtin(__builtin_amdgcn_global_load_async_to_lds_b32)
    typedef __attribute__((address_space(1))) int g_i32;   // __device__ int*
    typedef __attribute__((address_space(3))) int s_i32;   // __shared__ int*
    __builtin_amdgcn_global_load_async_to_lds_b32((g_i32*)(gp + 0), (s_i32*)&sm[t * 3 + 0], 0, 0);
    __builtin_amdgcn_global_load_async_to_lds_b32((g_i32*)(gp + 1), (s_i32*)&sm[t * 3 + 1], 0, 0);
    __builtin_amdgcn_global_load_async_to_lds_b32((g_i32*)(gp + 2), (s_i32*)&sm[t * 3 + 2], 0, 0);
#if __has_builtin(__builtin_amdgcn_s_wait_asynccnt)
    __builtin_amdgcn_s_wait_asynccnt(0);
#else
    asm volatile("s_wait_asynccnt 0" ::: "memory");
#endif
    // No barrier needed: each lane reads back only its own async-written slots,
    // and s_wait_asynccnt 0 guarantees this wave's async writes landed in LDS.
    v0 = sm[t * 3 + 0];
    v1 = sm[t * 3 + 1];
    v2 = sm[t * 3 + 2];
#else
    v0 = gp[0];
    v1 = gp[1];
    v2 = gp[2];
#endif

    int b  = idx / DHW;
    int r  = idx - b * DHW;
    int wq = r % WW;
    int tq = r / WW;
    int hq = tq % HH;
    int dq = tq / HH;

    // Torch code feeds (d,h,w)-displacements into grid_sample's (x,y,z)=(W,H,D)
    float ix = unnorm_clip((float)dq + v0);   // x -> W axis
    float iy = unnorm_clip((float)hq + v1);   // y -> H axis
    float iz = unnorm_clip((float)wq + v2);   // z -> D axis

    float x0f = floorf(ix), y0f = floorf(iy), z0f = floorf(iz);
    float wx = ix - x0f, wy = iy - y0f, wz = iz - z0f;
    int x0 = (int)x0f, y0 = (int)y0f, z0 = (int)z0f;
    int x1 = min(x0 + 1, WW - 1);
    int y1 = min(y0 + 1, HH - 1);
    int z1 = min(z0 + 1, DD - 1);
    float ux = 1.0f - wx, uy = 1.0f - wy, uz = 1.0f - wz;

    const float* vb = src + (size_t)b * DHW * 3u;
    float a0 = v0, a1 = v1, a2 = v2;   // out = v + warped

#define CORNER(zi, yi, xi, wt)                                                  \
    {                                                                           \
        const float* cp = vb + (size_t)((((zi) * HH + (yi)) * WW) + (xi)) * 3u; \
        float wv = (wt);                                                        \
        a0 = fmaf(wv, cp[0], a0);                                               \
        a1 = fmaf(wv, cp[1], a1);                                               \
        a2 = fmaf(wv, cp[2], a2);                                               \
    }

    CORNER(z0, y0, x0, uz * uy * ux);
    CORNER(z0, y0, x1, uz * uy * wx);
    CORNER(z0, y1, x0, uz * wy * ux);
    CORNER(z0, y1, x1, uz * wy * wx);
    CORNER(z1, y0, x0, wz * uy * ux);
    CORNER(z1, y0, x1, wz * uy * wx);
    CORNER(z1, y1, x0, wz * wy * ux);
    CORNER(z1, y1, x1, wz * wy * wx);
#undef CORNER

    if (soa_out) {
        float* o = dst + (size_t)b * 3u * DHW + r;
        o[0] = a0;
        o[DHW] = a1;
        o[2 * (size_t)DHW] = a2;
    } else {
        float* o = dst + (size_t)idx * 3u;
        o[0] = a0;
        o[1] = a1;
        o[2] = a2;
    }
}

extern "C" __global__ void __launch_bounds__(TPB)
saq_prescale(const float* __restrict__ vin, float* __restrict__ aos) {
    int idx = blockIdx.x * TPB + threadIdx.x;
    int b = idx / DHW;
    int r = idx - b * DHW;
    const float* p = vin + (size_t)b * 3u * DHW + r;
    float* q = aos + (size_t)idx * 3u;
    const float s = 0.0078125f;   // 2^-7
    q[0] = p[0] * s;
    q[1] = p[DHW] * s;
    q[2] = p[2 * (size_t)DHW] * s;
}

extern "C" void kernel_launch(void* const* d_in, const int* in_sizes, int n_in,
                              void* d_out, int out_size, void* d_ws, size_t ws_size,
                              hipStream_t stream) {
    (void)in_sizes; (void)n_in; (void)out_size; (void)ws_size;
    const float* v = (const float*)d_in[0];
    float* A = (float*)d_ws;    // AoS scratch (needs 98.3 MB)
    float* B = (float*)d_out;   // AoS intermediate; final step rewrites as SoA
    dim3 grid(NBLK), blk(TPB);

    saq_prescale<<<grid, blk, 0, stream>>>(v, A);          // v / 2^7, SoA->AoS
    saq_step<<<grid, blk, 0, stream>>>(A, B, 0);           // step 0
    saq_step<<<grid, blk, 0, stream>>>(B, A, 0);           // step 1
    saq_step<<<grid, blk, 0, stream>>>(A, B, 0);           // step 2
    saq_step<<<grid, blk, 0, stream>>>(B, A, 0);           // step 3
    saq_step<<<grid, blk, 0, stream>>>(A, B, 0);           // step 4
    saq_step<<<grid, blk, 0, stream>>>(B, A, 0);           // step 5
    saq_step<<<grid, blk, 0, stream>>>(A, B, 1);           // step 6 -> SoA out
}